// SparseGrid_67164698574947
// MI455X (gfx1250) — compile-verified
//
#include <hip/hip_runtime.h>
#include <cstdint>
#include <cstddef>

// ---------------------------------------------------------------------------
// Plenoxels-style volume renderer for MI455X (gfx1250, wave32).
//   grid: 128^3, density f32 + 27 SH f32 per voxel
//   16384 rays x 192 samples, one wave32 per ray, lane = sample-in-chunk.
// Memory-latency-bound kernel: wave-per-ray for 32-deep gather MLP,
// wave shuffles for the transmittance scan, WMMA f32 16x16x4 for the
// SH->rgb contraction on the matrix pipe, global_prefetch for look-ahead.
// ---------------------------------------------------------------------------

#define RESO        128
#define STEPSZ      0.5f
#define SIGMA_TH    1e-10f

typedef __attribute__((ext_vector_type(2))) float v2f;
typedef __attribute__((ext_vector_type(4))) float v4f;
typedef __attribute__((ext_vector_type(8))) float v8f;

// A-matrix value: Arep[row, J] = basis[J%9] when row == J/9 (J<27), else 0.
// J is compile-time constant after unrolling -> folds to one v_cndmask.
__device__ __forceinline__ float Aval(int row, int J, const float* b) {
  if (J >= 27) return 0.0f;
  const int c = J / 9;
  const int k = J % 9;
  return (row == c) ? b[k] : 0.0f;
}

// Guarded register-array read; j is constant after unrolling (dead branch folds).
__device__ __forceinline__ float sget(const float s[27], int j) {
  return (j < 27) ? s[(j < 27) ? j : 0] : 0.0f;
}

__device__ __forceinline__ int clamp126(int v) {
  return v < 0 ? 0 : (v > 126 ? 126 : v);
}

__global__ __launch_bounds__(256) void plenoxel_render(
    const float* __restrict__ origins, const float* __restrict__ dirs,
    const float* __restrict__ density, const float* __restrict__ sh,
    float* __restrict__ out) {
  const int lane = threadIdx.x & 31;                    // wave32: lane in wave
  const int ray  = blockIdx.x * (blockDim.x >> 5) + (threadIdx.x >> 5);
  const int row  = lane & 15;
  const bool hi  = lane >= 16;

  const float ox = origins[ray * 3 + 0];
  const float oy = origins[ray * 3 + 1];
  const float oz = origins[ray * 3 + 2];
  const float dx = dirs[ray * 3 + 0];
  const float dy = dirs[ray * 3 + 1];
  const float dz = dirs[ray * 3 + 2];

  // ---- degree-2 real SH basis (svox2 convention); uniform across the wave
  float b9[9];
  b9[0] = 0.28209479177387814f;
  b9[1] = -0.4886025119029199f * dy;
  b9[2] =  0.4886025119029199f * dz;
  b9[3] = -0.4886025119029199f * dx;
  b9[4] =  1.0925484305920792f * dx * dy;
  b9[5] = -1.0925484305920792f * dy * dz;
  b9[6] =  0.31539156525252005f * (2.0f * dz * dz - dx * dx - dy * dy);
  b9[7] = -1.0925484305920792f * dx * dz;
  b9[8] =  0.5462742152960396f * (dx * dx - dy * dy);

  // ---- WMMA A operand, built once per ray.
  // Arep is 16x28 (rows 0..2 = channels, rest zero; K padded 27->28).
  // 16x4 f32 A layout: lanes 0-15 hold K=j0,j0+1 (vgpr0,1); lanes 16-31 K=j0+2,j0+3.
  v2f Areg[7];
#pragma unroll
  for (int q = 0; q < 7; ++q) {
    const float a0lo = Aval(row, 4 * q + 0, b9);
    const float a1lo = Aval(row, 4 * q + 1, b9);
    const float a0hi = Aval(row, 4 * q + 2, b9);
    const float a1hi = Aval(row, 4 * q + 3, b9);
    Areg[q].x = hi ? a0hi : a0lo;
    Areg[q].y = hi ? a1hi : a1lo;
  }

  float acc0 = 0.0f, acc1 = 0.0f, acc2 = 0.0f;   // weighted color partials
  float carry = 0.0f;                             // cumulative log-transmittance

#pragma unroll 1
  for (int it = 0; it < 6; ++it) {                // 6 x 32 = 192 samples
    const float t  = ((float)(it * 32 + lane) + 0.5f) * STEPSZ;
    const float px = fmaf(t, dx, ox);
    const float py = fmaf(t, dy, oy);
    const float pz = fmaf(t, dz, oz);
    const bool inb = (px >= 0.0f) && (px <= 127.0f) &&
                     (py >= 0.0f) && (py <= 127.0f) &&
                     (pz >= 0.0f) && (pz <= 127.0f);

    const float fx0 = floorf(px), fy0 = floorf(py), fz0 = floorf(pz);
    const float fx = px - fx0, fy = py - fy0, fz = pz - fz0;
    const int ix = clamp126((int)fx0);
    const int iy = clamp126((int)fy0);
    const int iz = clamp126((int)fz0);

    // Prefetch next iteration's cell (16 voxels ahead on this lane's ray).
    {
      const float tn = t + 16.0f;
      const int jx = clamp126((int)floorf(fmaf(tn, dx, ox)));
      const int jy = clamp126((int)floorf(fmaf(tn, dy, oy)));
      const int jz = clamp126((int)floorf(fmaf(tn, dz, oz)));
      const int idxn = (jx << 14) | (jy << 7) | jz;
      __builtin_prefetch(sh + (size_t)idxn * 27, 0, 1);   // -> global_prefetch_b8
      __builtin_prefetch(density + idxn, 0, 1);
    }

    // ---- trilinear gather of density + 27 SH channels (8 corners)
    float s27[27];
#pragma unroll
    for (int k = 0; k < 27; ++k) s27[k] = 0.0f;
    float sig = 0.0f;

#pragma unroll
    for (int cx = 0; cx < 2; ++cx) {
      const float wx = cx ? fx : 1.0f - fx;
      const int bx = (ix + cx) << 14;
#pragma unroll
      for (int cy = 0; cy < 2; ++cy) {
        const float wxy = wx * (cy ? fy : 1.0f - fy);
        const int bxy = bx | ((iy + cy) << 7);
#pragma unroll
        for (int cz = 0; cz < 2; ++cz) {
          const float w = wxy * (cz ? fz : 1.0f - fz);
          const int idx = bxy | (iz + cz);
          sig = fmaf(w, density[idx], sig);
          const float* p = sh + (size_t)idx * 27;
#pragma unroll
          for (int v = 0; v < 6; ++v) {           // 6 x 16B wide loads
            v4f qv;
            __builtin_memcpy(&qv, p + 4 * v, sizeof(qv));
            s27[4 * v + 0] = fmaf(w, qv.x, s27[4 * v + 0]);
            s27[4 * v + 1] = fmaf(w, qv.y, s27[4 * v + 1]);
            s27[4 * v + 2] = fmaf(w, qv.z, s27[4 * v + 2]);
            s27[4 * v + 3] = fmaf(w, qv.w, s27[4 * v + 3]);
          }
          s27[24] = fmaf(w, p[24], s27[24]);
          s27[25] = fmaf(w, p[25], s27[25]);
          s27[26] = fmaf(w, p[26], s27[26]);
        }
      }
    }

    // ---- opacity + wave-level exclusive transmittance scan
    const float sigma = (inb && (sig > SIGMA_TH)) ? sig : 0.0f;
    const float la    = -sigma * STEPSZ;
    const float alpha = 1.0f - __expf(la);
    float incl = la;
#pragma unroll
    for (int d = 1; d < 32; d <<= 1) {
      const float u = __shfl_up(incl, d);
      incl += (lane >= d) ? u : 0.0f;
    }
    const float w_i = __expf(carry + (incl - la)) * alpha;  // weight of my sample
    carry += __shfl(incl, 31);                              // chunk total (uniform)

    // ---- SH -> rgb on the matrix pipe: D[3x16] = Arep[3x27] * shv[27x16]
    // Two 16-sample groups; 7 chained K=4 f32 WMMAs each.
    // B 4x16 layout assumed symmetric to A: vgpr0 = rows K=j0 (lanes 0-15) /
    // K=j0+2 (lanes 16-31); vgpr1 = K=j0+1 / K=j0+3.
#pragma unroll
    for (int g = 0; g < 2; ++g) {
      const int src = g * 16 + (lane & 15);       // sample feeding my column
      v8f D = {0, 0, 0, 0, 0, 0, 0, 0};
#pragma unroll
      for (int q = 0; q < 7; ++q) {
        const int j0 = 4 * q;
        const float u0 = __shfl(sget(s27, j0 + 0), src);
        const float u1 = __shfl(sget(s27, j0 + 1), src);
        const float u2 = __shfl(sget(s27, j0 + 2), src);
        const float u3 = __shfl(sget(s27, j0 + 3), src);
        v2f B;
        B.x = hi ? u2 : u0;
        B.y = hi ? u3 : u1;
        D = __builtin_amdgcn_wmma_f32_16x16x4_f32(
            false, Areg[q], false, B, (short)0, D, false, false);
      }
      // D vgpr c, lanes 0-15 = rgb channel c of sample (g*16 + lane).
      // Lanes 16-31 hold rows 8+c (all-zero A rows) -> masked by wsel=0.
      float wsel;
      if (g == 0) {
        wsel = hi ? 0.0f : w_i;
      } else {
        const float wh = __shfl(w_i, (lane & 15) + 16);
        wsel = hi ? 0.0f : wh;
      }
      acc0 = fmaf(fmaxf(D[0] + 0.5f, 0.0f), wsel, acc0);
      acc1 = fmaf(fmaxf(D[1] + 0.5f, 0.0f), wsel, acc1);
      acc2 = fmaf(fmaxf(D[2] + 0.5f, 0.0f), wsel, acc2);
    }
  }

  // ---- reduce weighted color across the wave, add empty-space term
#pragma unroll
  for (int d = 16; d >= 1; d >>= 1) {
    acc0 += __shfl_down(acc0, d);
    acc1 += __shfl_down(acc1, d);
    acc2 += __shfl_down(acc2, d);
  }
  const float Tfin = __expf(carry);               // uniform across wave
  if (lane == 0) {
    out[ray * 3 + 0] = acc0 + Tfin;
    out[ray * 3 + 1] = acc1 + Tfin;
    out[ray * 3 + 2] = acc2 + Tfin;
  }
}

extern "C" void kernel_launch(void* const* d_in, const int* in_sizes, int n_in,
                              void* d_out, int out_size, void* d_ws, size_t ws_size,
                              hipStream_t stream) {
  const float* origins = (const float*)d_in[0];   // [N,3]
  const float* dirs    = (const float*)d_in[1];   // [N,3]
  const float* density = (const float*)d_in[2];   // [128^3]
  const float* sh      = (const float*)d_in[3];   // [128^3, 27]
  float* out = (float*)d_out;                     // [N,3]

  const int nrays = in_sizes[0] / 3;              // 16384
  // one wave32 per ray, 8 rays per 256-thread block
  const int blocks = nrays / 8;
  plenoxel_render<<<blocks, 256, 0, stream>>>(origins, dirs, density, sh, out);
}